// PyTorchPoseOptimizer_14920716386818
// MI455X (gfx1250) — compile-verified
//
#include <hip/hip_runtime.h>
#include <math.h>

typedef __attribute__((ext_vector_type(2))) float v2f;
typedef __attribute__((ext_vector_type(8))) float v8f;

#define FORWARD_W 1.0f
#define BACKWARD_W 1.0f
#define NORMAL_W 0.1f
#define EPSV 1e-6f

// ---------------------------------------------------------------------------
// Rotation 6D -> 3x3 matrix (Gram-Schmidt), computed redundantly per thread
// (only ~40 flops; avoids an extra kernel / sync).
// ---------------------------------------------------------------------------
__device__ inline void rot6d_to_R(const float* __restrict__ r6, float R[3][3]) {
    float a1x = r6[0], a1y = r6[1], a1z = r6[2];
    float a2x = r6[3], a2y = r6[4], a2z = r6[5];
    float n1 = sqrtf(a1x * a1x + a1y * a1y + a1z * a1z);
    float b1x = a1x / n1, b1y = a1y / n1, b1z = a1z / n1;
    float d = b1x * a2x + b1y * a2y + b1z * a2z;
    float b2x = a2x - d * b1x, b2y = a2y - d * b1y, b2z = a2z - d * b1z;
    float n2 = sqrtf(b2x * b2x + b2y * b2y + b2z * b2z);
    b2x /= n2; b2y /= n2; b2z /= n2;
    float b3x = b1y * b2z - b1z * b2y;
    float b3y = b1z * b2x - b1x * b2z;
    float b3z = b1x * b2y - b1y * b2x;
    // R = stack([b1,b2,b3], axis=1): row i = (b1[i], b2[i], b3[i])
    R[0][0] = b1x; R[0][1] = b2x; R[0][2] = b3x;
    R[1][0] = b1y; R[1][1] = b2y; R[1][2] = b3y;
    R[2][0] = b1z; R[2][1] = b2z; R[2][2] = b3z;
}

// ---------------------------------------------------------------------------
// Prep: transform source points/normals, pad both clouds to stride-4 (4th
// component = 0 so WMMA K=4 works), compute squared norms, init u64 mins.
// ---------------------------------------------------------------------------
__global__ void prep_kernel(const float* __restrict__ sp, const float* __restrict__ tp,
                            const float* __restrict__ sn, const float* __restrict__ trans,
                            const float* __restrict__ r6, const float* __restrict__ scl,
                            float* __restrict__ srcP4, float* __restrict__ tgtP4,
                            float* __restrict__ srcN4, float* __restrict__ snorm,
                            float* __restrict__ tnorm,
                            unsigned long long* __restrict__ rmS,
                            unsigned long long* __restrict__ rmT, int N, int M) {
    int i = blockIdx.x * blockDim.x + threadIdx.x;
    float R[3][3];
    rot6d_to_R(r6, R);
    if (i < N) {
        float w0 = sp[i * 3 + 0] * scl[0];
        float w1 = sp[i * 3 + 1] * scl[1];
        float w2 = sp[i * 3 + 2] * scl[2];
        float sx = R[0][0] * w0 + R[0][1] * w1 + R[0][2] * w2 + trans[0];
        float sy = R[1][0] * w0 + R[1][1] * w1 + R[1][2] * w2 + trans[1];
        float sz = R[2][0] * w0 + R[2][1] * w1 + R[2][2] * w2 + trans[2];
        srcP4[i * 4 + 0] = sx; srcP4[i * 4 + 1] = sy;
        srcP4[i * 4 + 2] = sz; srcP4[i * 4 + 3] = 0.0f;
        snorm[i] = sx * sx + sy * sy + sz * sz;
        float u0 = sn[i * 3 + 0], u1 = sn[i * 3 + 1], u2 = sn[i * 3 + 2];
        srcN4[i * 4 + 0] = R[0][0] * u0 + R[0][1] * u1 + R[0][2] * u2;
        srcN4[i * 4 + 1] = R[1][0] * u0 + R[1][1] * u1 + R[1][2] * u2;
        srcN4[i * 4 + 2] = R[2][0] * u0 + R[2][1] * u1 + R[2][2] * u2;
        srcN4[i * 4 + 3] = 0.0f;
        rmS[i] = ~0ull;
    }
    if (i < M) {
        float tx = tp[i * 3 + 0], ty = tp[i * 3 + 1], tz = tp[i * 3 + 2];
        tgtP4[i * 4 + 0] = tx; tgtP4[i * 4 + 1] = ty;
        tgtP4[i * 4 + 2] = tz; tgtP4[i * 4 + 3] = 0.0f;
        tnorm[i] = tx * tx + ty * ty + tz * tz;
        rmT[i] = ~0ull;
    }
}

// ---------------------------------------------------------------------------
// Core: each wave owns a 16-row strip of A, sweeps all columns of B in 16-wide
// tiles. V_WMMA_F32_16X16X4_F32 computes D = (-2*A) x B + (|a|^2 + |b|^2)
// = the d2 tile directly. Row-min+argmin tracked as packed u64 keys.
//   A4/B4: stride-4 padded points; An/Bn: squared norms.
//   rowmin[i] = (bits(min d2) << 32) | argmin  over all B columns.
// ---------------------------------------------------------------------------
__global__ void chamfer_strip_kernel(const float* __restrict__ A4, const float* __restrict__ B4,
                                     const float* __restrict__ An, const float* __restrict__ Bn,
                                     unsigned long long* __restrict__ rowmin, int nB) {
    const int lane = threadIdx.x & 31;
    const int wave = threadIdx.x >> 5;
    const int strip = blockIdx.x * (blockDim.x >> 5) + wave;
    const int m_lo = lane & 15;       // row (A) / col (B) within tile
    const int hi = lane >> 4;         // half-wave select
    const int kb = hi << 1;           // K components {0,1} or {2,3}
    const int rowbase = strip * 16;

    // A operand: loop-invariant, pre-scaled by -2 so WMMA emits d2 directly.
    v2f a;
    a.x = -2.0f * A4[(rowbase + m_lo) * 4 + kb + 0];
    a.y = -2.0f * A4[(rowbase + m_lo) * 4 + kb + 1];

    // |a_m|^2 for this lane's 8 accumulator rows: m_r = r + 8*hi
    float sa[8];
#pragma unroll
    for (int r = 0; r < 8; ++r) sa[r] = An[rowbase + r + 8 * hi];

    unsigned long long best[8];
#pragma unroll
    for (int r = 0; r < 8; ++r) best[r] = ~0ull;

    for (int tile = 0; tile < nB; tile += 16) {
        const int n = tile + m_lo;    // this lane's B column
        v2f b;
        b.x = B4[n * 4 + kb + 0];
        b.y = B4[n * 4 + kb + 1];
        const float tb = Bn[n];
        v8f c;
#pragma unroll
        for (int r = 0; r < 8; ++r) c[r] = sa[r] + tb;
        // D = (-2A)xB + C : 16x16 tile of squared distances
        v8f d = __builtin_amdgcn_wmma_f32_16x16x4_f32(
            false, a, false, b, (short)0, c, false, false);
#pragma unroll
        for (int r = 0; r < 8; ++r) {
            float v = fmaxf(d[r], 0.0f);  // d2 = max(d2, 0): bits now monotone
            unsigned long long key =
                ((unsigned long long)__float_as_uint(v) << 32) | (unsigned int)n;
            best[r] = (key < best[r]) ? key : best[r];
        }
    }

    // Butterfly min over the 16 lanes of each half-wave, then one store/row.
#pragma unroll
    for (int r = 0; r < 8; ++r) {
        unsigned long long k = best[r];
        for (int mask = 8; mask >= 1; mask >>= 1) {
            unsigned long long o = __shfl_xor(k, mask, 16);
            k = (o < k) ? o : k;
        }
        if (m_lo == 0) rowmin[rowbase + r + 8 * hi] = k;
    }
}

// ---------------------------------------------------------------------------
// Epilogue: per-point distance + normal-cosine contributions, deterministic
// block partial sums (no float atomics).
// ---------------------------------------------------------------------------
__global__ void finalize_kernel(const unsigned long long* __restrict__ rmS,
                                const unsigned long long* __restrict__ rmT,
                                const float* __restrict__ srcN4,
                                const float* __restrict__ tn,
                                float* __restrict__ partial, int N, int M) {
    int i = blockIdx.x * blockDim.x + threadIdx.x;
    float contrib = 0.0f;
    if (i < N) {
        unsigned long long ks = rmS[i];
        float ds = __uint_as_float((unsigned int)(ks >> 32));
        int js = (int)(unsigned int)(ks & 0xffffffffu);
        float ax = srcN4[i * 4 + 0], ay = srcN4[i * 4 + 1], az = srcN4[i * 4 + 2];
        float bx = tn[js * 3 + 0], by = tn[js * 3 + 1], bz = tn[js * 3 + 2];
        float na = fmaxf(sqrtf(ax * ax + ay * ay + az * az), EPSV);
        float nb = fmaxf(sqrtf(bx * bx + by * by + bz * bz), EPSV);
        float cns = 1.0f - fabsf((ax * bx + ay * by + az * bz) / (na * nb));
        contrib += FORWARD_W * ds * (1.0f / (float)N) + NORMAL_W * cns * (1.0f / (float)N);
    }
    if (i < M) {
        unsigned long long kt = rmT[i];
        float dt = __uint_as_float((unsigned int)(kt >> 32));
        int jt = (int)(unsigned int)(kt & 0xffffffffu);
        float cx = tn[i * 3 + 0], cy = tn[i * 3 + 1], cz = tn[i * 3 + 2];
        float ex = srcN4[jt * 4 + 0], ey = srcN4[jt * 4 + 1], ez = srcN4[jt * 4 + 2];
        float nc = fmaxf(sqrtf(cx * cx + cy * cy + cz * cz), EPSV);
        float ne = fmaxf(sqrtf(ex * ex + ey * ey + ez * ez), EPSV);
        float cnt = 1.0f - fabsf((cx * ex + cy * ey + cz * ez) / (nc * ne));
        contrib += BACKWARD_W * dt * (1.0f / (float)M) + NORMAL_W * cnt * (1.0f / (float)M);
    }
    // in-block tree reduction (deterministic)
    __shared__ float sdata[8];
    for (int off = 16; off >= 1; off >>= 1)
        contrib += __shfl_down(contrib, off, 32);
    int lane = threadIdx.x & 31, wid = threadIdx.x >> 5;
    if (lane == 0) sdata[wid] = contrib;
    __syncthreads();
    if (wid == 0) {
        float v = (lane < (int)(blockDim.x >> 5)) ? sdata[lane] : 0.0f;
        for (int off = 4; off >= 1; off >>= 1)
            v += __shfl_down(v, off, 32);
        if (lane == 0) partial[blockIdx.x] = v;
    }
}

__global__ void sum_kernel(const float* __restrict__ partial, float* __restrict__ out, int n) {
    float v = ((int)threadIdx.x < n) ? partial[threadIdx.x] : 0.0f;
    for (int off = 16; off >= 1; off >>= 1)
        v += __shfl_down(v, off, 32);
    if (threadIdx.x == 0) out[0] = v;
}

// ---------------------------------------------------------------------------
// Launch: prep -> s2t strip-min -> t2s strip-min (operands swapped) ->
// epilogue -> final sum.  All on `stream`, no atomics, deterministic.
// ---------------------------------------------------------------------------
extern "C" void kernel_launch(void* const* d_in, const int* in_sizes, int n_in,
                              void* d_out, int out_size, void* d_ws, size_t ws_size,
                              hipStream_t stream) {
    const float* sp    = (const float*)d_in[0];
    const float* tp    = (const float*)d_in[1];
    const float* sn    = (const float*)d_in[2];
    const float* tn    = (const float*)d_in[3];
    const float* trans = (const float*)d_in[4];
    const float* r6    = (const float*)d_in[5];
    const float* scl   = (const float*)d_in[6];
    const int N = in_sizes[0] / 3;   // 8192
    const int M = in_sizes[1] / 3;   // 8192

    // Workspace layout (all offsets 8-byte aligned for N,M multiples of 16)
    float* ws = (float*)d_ws;
    float* srcP4 = ws;                 // N*4
    float* tgtP4 = srcP4 + (size_t)N * 4;   // M*4
    float* srcN4 = tgtP4 + (size_t)M * 4;   // N*4
    float* snorm = srcN4 + (size_t)N * 4;   // N
    float* tnorm = snorm + N;               // M
    unsigned long long* rmS = (unsigned long long*)(tnorm + M);  // N u64
    unsigned long long* rmT = rmS + N;                           // M u64
    float* partial = (float*)(rmT + M);                          // <= 64 floats

    const int prep_n = (N > M ? N : M);
    prep_kernel<<<(prep_n + 255) / 256, 256, 0, stream>>>(
        sp, tp, sn, trans, r6, scl, srcP4, tgtP4, srcN4, snorm, tnorm, rmS, rmT, N, M);

    // 8 waves / block; one wave per 16-row strip.
    const int stripsS = N / 16;
    const int stripsT = M / 16;
    chamfer_strip_kernel<<<stripsS / 8, 256, 0, stream>>>(srcP4, tgtP4, snorm, tnorm, rmS, M);
    chamfer_strip_kernel<<<stripsT / 8, 256, 0, stream>>>(tgtP4, srcP4, tnorm, snorm, rmT, N);

    const int fblocks = (prep_n + 255) / 256;  // 32 for 8192
    finalize_kernel<<<fblocks, 256, 0, stream>>>(rmS, rmT, srcN4, tn, partial, N, M);
    sum_kernel<<<1, 32, 0, stream>>>(partial, (float*)d_out, fblocks);
}